// Mixed_79697413144643
// MI455X (gfx1250) — compile-verified
//
#include <hip/hip_runtime.h>
#include <math.h>

#define N_NODES 50000
#define N_EDGES 800000
#define N_REL   4
#define F_IN    128
#define H1      64
#define HEADS   4
#define D_H     32
#define HD      128   /* HEADS*D_H */
#define N_GRAPHS 64

typedef __attribute__((ext_vector_type(2))) float v2f;
typedef __attribute__((ext_vector_type(8))) float v8f;
typedef __attribute__((ext_vector_type(4))) int   v4i;

// ---------- async global->LDS staging (CDNA5), with portable fallback ----------
#if defined(__has_builtin)
#  if __has_builtin(__builtin_amdgcn_global_load_async_to_lds_b128)
#    define ASYNC_LDS 1
#  endif
#endif
#ifndef ASYNC_LDS
#  define ASYNC_LDS 0
#endif

typedef __attribute__((address_space(1))) v4i* gptr_v4i;
typedef __attribute__((address_space(3))) v4i* lptr_v4i;

__device__ __forceinline__ void async_copy_b128(const float* g, float* l) {
#if ASYNC_LDS
  __builtin_amdgcn_global_load_async_to_lds_b128((gptr_v4i)g, (lptr_v4i)l, 0, 0);
#else
  *(float4*)l = *(const float4*)g;
#endif
}

__device__ __forceinline__ void async_join() {
#if ASYNC_LDS
#  if __has_builtin(__builtin_amdgcn_s_wait_asynccnt)
  __builtin_amdgcn_s_wait_asynccnt(0);
#  else
  asm volatile("s_wait_asynccnt 0" ::: "memory");
#  endif
#endif
}

// ---- order-preserving float<->uint mapping for atomicMax-based segment_max ----
__device__ __forceinline__ unsigned encOrd(float f) {
  unsigned b = __float_as_uint(f);
  return (b & 0x80000000u) ? ~b : (b ^ 0x80000000u);
}
__device__ __forceinline__ float decOrd(unsigned k) {
  unsigned b = (k & 0x80000000u) ? (k ^ 0x80000000u) : ~k;
  return __uint_as_float(b);
}
#define ENC_NEG_INF 0x007FFFFFu   /* encOrd(-inf) */

// ---- fp32 WMMA 16x16 tile: C += A[16xK] * B[Kx16] from LDS, K%4==0 ----
__device__ __forceinline__ v8f wmma_f32_tile(const float* As, int asStride,
                                             const float* Bs, int bsStride,
                                             int K, int nOff, int lane) {
  const int m  = lane & 15;          // A row
  const int kb = (lane >> 4) << 1;   // K pair select (lanes 16-31 -> K+2,K+3)
  const int n  = (lane & 15) + nOff; // B col
  v8f acc = {0.f, 0.f, 0.f, 0.f, 0.f, 0.f, 0.f, 0.f};
  for (int kk = 0; kk < K; kk += 4) {
    v2f a, b;
    a.x = As[m * asStride + kk + kb];
    a.y = As[m * asStride + kk + kb + 1];
    b.x = Bs[(kk + kb) * bsStride + n];
    b.y = Bs[(kk + kb + 1) * bsStride + n];
    acc = __builtin_amdgcn_wmma_f32_16x16x4_f32(false, a, false, b,
                                                (short)0, acc, false, false);
  }
  return acc;
}

// ------------------------- init kernels -------------------------
__global__ void k_zero_f32(float* p, int n) {
  int i = blockIdx.x * 256 + threadIdx.x;
  if (i < n) p[i] = 0.f;
}
__global__ void k_fill_u32(unsigned* p, unsigned v, int n) {
  int i = blockIdx.x * 256 + threadIdx.x;
  if (i < n) p[i] = v;
}

// ------------- GEMM1: [N,128] x [128, 64*(1+4)] via WMMA ---------------
// grid (N/16, 5): y=0 -> W_root (+bias) into rootbuf; y=1..4 -> W_rel[r] into ybuf
__global__ void k_rgcn_gemm(const float* __restrict__ x,
                            const float* __restrict__ Wroot,
                            const float* __restrict__ Wrel,
                            const float* __restrict__ brgcn,
                            float* __restrict__ rootbuf,
                            float* __restrict__ ybuf) {
  __shared__ __align__(16) float As[16 * 132];   // stride 132: 16B-aligned rows, conflict-free
  __shared__ __align__(16) float Bs[F_IN * H1];
  const int tid  = threadIdx.x;  // 128
  const int row0 = blockIdx.x * 16;
  const int cg   = blockIdx.y;
  // A tile: 16 rows x 128 floats, b128 chunks (8 threads per row, 4 chunks each)
  {
    const int r = tid >> 3;
    for (int s = 0; s < 4; ++s) {
      const int c = ((tid & 7) + s * 8) * 4;
      async_copy_b128(&x[(size_t)(row0 + r) * F_IN + c], &As[r * 132 + c]);
    }
  }
  // B slab: contiguous 128x64 floats = 2048 b128 chunks
  const float* Bsrc = (cg == 0) ? Wroot : (Wrel + (size_t)(cg - 1) * F_IN * H1);
  for (int s = 0; s < 16; ++s) {
    const int q = (tid + s * 128) * 4;
    async_copy_b128(Bsrc + q, &Bs[q]);
  }
  async_join();
  __syncthreads();
  const int wave = tid >> 5, lane = tid & 31;
  v8f acc = wmma_f32_tile(As, 132, Bs, H1, F_IN, wave * 16, lane);
  const int cl = wave * 16 + (lane & 15);
  const int rh = (lane >> 4) * 8;
  for (int i = 0; i < 8; ++i) {
    int r = row0 + rh + i;
    float v = acc[i];
    if (cg == 0) rootbuf[(size_t)r * H1 + cl] = v + brgcn[cl];
    else         ybuf[(size_t)r * (N_REL * H1) + (cg - 1) * H1 + cl] = v;
  }
}

// ---- RGCN edge scatter of transformed features: 64 lanes per edge ----
__global__ void k_rgcn_scatter(const float* __restrict__ ybuf,
                               const int* __restrict__ ei,
                               const int* __restrict__ et,
                               float* __restrict__ acc,
                               float* __restrict__ cnt) {
  int e = blockIdx.x * 4 + (threadIdx.x >> 6);
  int j = threadIdx.x & 63;
  if (e >= N_EDGES) return;
  int src = ei[e], dst = ei[N_EDGES + e], r = et[e];
  const float* yp = ybuf + (size_t)src * (N_REL * H1) + r * H1;
  __builtin_prefetch(yp, 0, 0);
  atomicAdd(&acc[((size_t)r * N_NODES + dst) * H1 + j], yp[j]);
  if (j == 0) atomicAdd(&cnt[(size_t)r * N_NODES + dst], 1.0f);
}

// ---- combine: h = relu(root + sum_r acc_r / max(cnt_r,1)) ----
__global__ void k_rgcn_combine(const float* __restrict__ rootbuf,
                               const float* __restrict__ acc,
                               const float* __restrict__ cnt,
                               float* __restrict__ h) {
  int i = blockIdx.x * 256 + threadIdx.x;
  if (i >= N_NODES * H1) return;
  int n = i >> 6, c = i & 63;
  float v = rootbuf[i];
#pragma unroll
  for (int r = 0; r < N_REL; ++r) {
    float cr = cnt[(size_t)r * N_NODES + n];
    v += acc[((size_t)r * N_NODES + n) * H1 + c] / fmaxf(cr, 1.0f);
  }
  h[i] = fmaxf(v, 0.0f);
}

// ------------- GEMM2: g = h[N,64] x W_gat[64,128] via WMMA -------------
__global__ void k_gat_gemm(const float* __restrict__ h,
                           const float* __restrict__ Wgat,
                           float* __restrict__ g) {
  __shared__ __align__(16) float As[16 * 68];    // stride 68: aligned + conflict-free
  __shared__ __align__(16) float Bs[H1 * 64];
  const int tid  = threadIdx.x;  // 128
  const int row0 = blockIdx.x * 16;
  const int cg   = blockIdx.y;   // 0..1 -> 64-col halves
  // A tile: 16 rows x 64 floats = 256 chunks, 2 per thread
  for (int s = 0; s < 2; ++s) {
    const int q = tid + s * 128;
    const int r = q >> 4, c = (q & 15) * 4;
    async_copy_b128(&h[(size_t)(row0 + r) * H1 + c], &As[r * 68 + c]);
  }
  // B slab: 64 rows x 64 floats (strided source rows) = 1024 chunks, 8 per thread
  for (int s = 0; s < 8; ++s) {
    const int q = tid + s * 128;
    const int k = q >> 4, c = (q & 15) * 4;
    async_copy_b128(&Wgat[(size_t)k * HD + cg * 64 + c], &Bs[k * 64 + c]);
  }
  async_join();
  __syncthreads();
  const int wave = tid >> 5, lane = tid & 31;
  v8f acc = wmma_f32_tile(As, 68, Bs, 64, H1, wave * 16, lane);
  const int cl = cg * 64 + wave * 16 + (lane & 15);
  const int rh = (lane >> 4) * 8;
  for (int i = 0; i < 8; ++i)
    g[(size_t)(row0 + rh + i) * HD + cl] = acc[i];
}

// ---- per-node per-head attention scores ----
__global__ void k_att_scores(const float* __restrict__ g,
                             const float* __restrict__ attS,
                             const float* __restrict__ attD,
                             float* __restrict__ a_s, float* __restrict__ a_d) {
  int t = blockIdx.x * 256 + threadIdx.x;
  if (t >= N_NODES * HEADS) return;
  int n = t >> 2, hh = t & 3;
  const float* gp = g + (size_t)n * HD + hh * D_H;
  float ss = 0.f, sd = 0.f;
  for (int d = 0; d < D_H; ++d) {
    float gv = gp[d];
    ss += gv * attS[hh * D_H + d];
    sd += gv * attD[hh * D_H + d];
  }
  a_s[t] = ss;
  a_d[t] = sd;
}

// ---- edge pass 1: segment_max of leaky_relu(a_s[src]+a_d[dst]) ----
__global__ void k_att_max(const int* __restrict__ ei,
                          const float* __restrict__ a_s,
                          const float* __restrict__ a_d,
                          unsigned* __restrict__ attkey) {
  int e = blockIdx.x * 256 + threadIdx.x;
  if (e >= N_EDGES) return;
  int src = ei[e], dst = ei[N_EDGES + e];
#pragma unroll
  for (int hh = 0; hh < HEADS; ++hh) {
    float v = a_s[src * 4 + hh] + a_d[dst * 4 + hh];
    v = v > 0.f ? v : 0.2f * v;
    atomicMax(&attkey[dst * 4 + hh], encOrd(v));
  }
}

// ---- edge pass 2: ex = exp(e - m[dst]); num += ex*g[src]; den += ex ----
__global__ void k_att_accum(const int* __restrict__ ei,
                            const float* __restrict__ a_s,
                            const float* __restrict__ a_d,
                            const unsigned* __restrict__ attkey,
                            const float* __restrict__ g,
                            float* __restrict__ num, float* __restrict__ den) {
  int e = blockIdx.x * 2 + (threadIdx.x >> 7);
  int j = threadIdx.x & 127;
  if (e >= N_EDGES) return;
  int hh = j >> 5;
  int src = ei[e], dst = ei[N_EDGES + e];
  float v = a_s[src * 4 + hh] + a_d[dst * 4 + hh];
  v = v > 0.f ? v : 0.2f * v;
  float m  = decOrd(attkey[dst * 4 + hh]);
  float ex = expf(v - m);
  float gv = g[(size_t)src * HD + j];
  atomicAdd(&num[(size_t)dst * HD + j], ex * gv);
  if ((j & 31) == 0) atomicAdd(&den[dst * 4 + hh], ex);
}

// ---- o = leaky( num/den + b_gat , 0.01) ; den==0 -> 0 (no in-edges) ----
__global__ void k_gat_final(const float* __restrict__ num,
                            const float* __restrict__ den,
                            const float* __restrict__ bgat,
                            float* __restrict__ o) {
  int i = blockIdx.x * 256 + threadIdx.x;
  if (i >= N_NODES * HD) return;
  int n = i >> 7, j = i & 127, hh = j >> 5;
  float d = den[n * 4 + hh];
  float v = (d > 0.f) ? num[i] / d : 0.f;
  v += bgat[j];
  o[i] = v > 0.f ? v : 0.01f * v;
}

// ---- GEMM3 fused with leaky + per-graph atomic max pool (1 wave/block) ----
__global__ void k_dense_pool(const float* __restrict__ o,
                             const float* __restrict__ W1,
                             const float* __restrict__ b1,
                             const int* __restrict__ batch,
                             unsigned* __restrict__ poolkey) {
  __shared__ __align__(16) float As[16 * 132];
  __shared__ __align__(16) float Bs[HD * 16];
  const int tid  = threadIdx.x;  // 32
  const int row0 = blockIdx.x * 16;
  // A tile: 16 rows x 128 floats = 512 chunks, 16 per thread
  for (int s = 0; s < 16; ++s) {
    const int q = tid + s * 32;
    const int r = q >> 5, c = (q & 31) * 4;
    async_copy_b128(&o[(size_t)(row0 + r) * HD + c], &As[r * 132 + c]);
  }
  // B slab: contiguous 128x16 floats = 512 chunks, 16 per thread
  for (int s = 0; s < 16; ++s) {
    const int q = (tid + s * 32) * 4;
    async_copy_b128(W1 + q, &Bs[q]);
  }
  async_join();
  __syncthreads();
  v8f acc = wmma_f32_tile(As, 132, Bs, 16, HD, 0, tid);
  const int cl = tid & 15;
  const int rh = (tid >> 4) * 8;
  for (int i = 0; i < 8; ++i) {
    int r = row0 + rh + i;
    float v = acc[i] + b1[cl];
    v = v > 0.f ? v : 0.01f * v;
    atomicMax(&poolkey[batch[r] * 16 + cl], encOrd(v));
  }
}

// ---- y[g] = p[g] @ W2 + b2 ----
__global__ void k_out(const unsigned* __restrict__ poolkey,
                      const float* __restrict__ W2,
                      const float* __restrict__ b2,
                      float* __restrict__ y) {
  int gidx = threadIdx.x;
  if (gidx >= N_GRAPHS) return;
  float s = b2[0];
  for (int c = 0; c < 16; ++c) s += decOrd(poolkey[gidx * 16 + c]) * W2[c];
  y[gidx] = s;
}

extern "C" void kernel_launch(void* const* d_in, const int* in_sizes, int n_in,
                              void* d_out, int out_size, void* d_ws, size_t ws_size,
                              hipStream_t stream) {
  (void)in_sizes; (void)n_in; (void)out_size; (void)ws_size;
  const float* x     = (const float*)d_in[0];
  const int*   ei    = (const int*)d_in[1];
  const int*   et    = (const int*)d_in[2];
  const int*   batch = (const int*)d_in[3];
  const float* Wrel  = (const float*)d_in[4];
  const float* Wroot = (const float*)d_in[5];
  const float* brgcn = (const float*)d_in[6];
  const float* Wgat  = (const float*)d_in[7];
  const float* attS  = (const float*)d_in[8];
  const float* attD  = (const float*)d_in[9];
  const float* bgat  = (const float*)d_in[10];
  const float* W1    = (const float*)d_in[11];
  const float* b1    = (const float*)d_in[12];
  const float* W2    = (const float*)d_in[13];
  const float* b2    = (const float*)d_in[14];

  char* ws = (char*)d_ws;
  size_t off = 0;
  auto alloc = [&](size_t bytes) -> void* {
    void* p = ws + off;
    off += (bytes + 255) & ~(size_t)255;
    return p;
  };
  // region A: ybuf[N,256]; reused later as num[N,128] + obuf[N,128]
  float* ybuf = (float*)alloc((size_t)N_NODES * N_REL * H1 * 4);
  float* num  = ybuf;
  float* obuf = ybuf + (size_t)N_NODES * HD;
  // region B: acc[4,N,64]; reused later as gbuf[N,128]
  float* accb = (float*)alloc((size_t)N_REL * N_NODES * H1 * 4);
  float* gbuf = accb;
  float* rootbuf   = (float*)alloc((size_t)N_NODES * H1 * 4);
  float* hbuf      = (float*)alloc((size_t)N_NODES * H1 * 4);
  float* cnt       = (float*)alloc((size_t)N_REL * N_NODES * 4);
  float* a_s       = (float*)alloc((size_t)N_NODES * HEADS * 4);
  float* a_d       = (float*)alloc((size_t)N_NODES * HEADS * 4);
  unsigned* attkey = (unsigned*)alloc((size_t)N_NODES * HEADS * 4);
  float* den       = (float*)alloc((size_t)N_NODES * HEADS * 4);
  unsigned* poolkey = (unsigned*)alloc((size_t)N_GRAPHS * 16 * 4);

  // --- init (re-done every call: deterministic under graph replay) ---
  const int nAcc = N_REL * N_NODES * H1;
  k_zero_f32<<<(nAcc + 255) / 256, 256, 0, stream>>>(accb, nAcc);
  k_zero_f32<<<(N_REL * N_NODES + 255) / 256, 256, 0, stream>>>(cnt, N_REL * N_NODES);
  k_fill_u32<<<(N_NODES * HEADS + 255) / 256, 256, 0, stream>>>(attkey, ENC_NEG_INF, N_NODES * HEADS);
  k_fill_u32<<<(N_GRAPHS * 16 + 255) / 256, 256, 0, stream>>>(poolkey, ENC_NEG_INF, N_GRAPHS * 16);

  // --- RGCN ---
  dim3 g1(N_NODES / 16, 1 + N_REL);
  k_rgcn_gemm<<<g1, 128, 0, stream>>>(x, Wroot, Wrel, brgcn, rootbuf, ybuf);
  k_rgcn_scatter<<<N_EDGES / 4, 256, 0, stream>>>(ybuf, ei, et, accb, cnt);
  k_rgcn_combine<<<(N_NODES * H1 + 255) / 256, 256, 0, stream>>>(rootbuf, accb, cnt, hbuf);

  // zero num/den AFTER scatter (num aliases ybuf, which scatter reads)
  k_zero_f32<<<(N_NODES * HD + 255) / 256, 256, 0, stream>>>(num, N_NODES * HD);
  k_zero_f32<<<(N_NODES * HEADS + 255) / 256, 256, 0, stream>>>(den, N_NODES * HEADS);

  // --- GAT ---
  dim3 g2(N_NODES / 16, 2);
  k_gat_gemm<<<g2, 128, 0, stream>>>(hbuf, Wgat, gbuf);
  k_att_scores<<<(N_NODES * HEADS + 255) / 256, 256, 0, stream>>>(gbuf, attS, attD, a_s, a_d);
  k_att_max<<<(N_EDGES + 255) / 256, 256, 0, stream>>>(ei, a_s, a_d, attkey);
  k_att_accum<<<N_EDGES / 2, 256, 0, stream>>>(ei, a_s, a_d, attkey, gbuf, num, den);
  k_gat_final<<<(N_NODES * HD + 255) / 256, 256, 0, stream>>>(num, den, bgat, obuf);

  // --- dense1 + pool fused, then output ---
  k_dense_pool<<<N_NODES / 16, 32, 0, stream>>>(obuf, W1, b1, batch, poolkey);
  k_out<<<1, 64, 0, stream>>>(poolkey, W2, b2, (float*)d_out);
}